// SimpleSAN_29506425323977
// MI455X (gfx1250) — compile-verified
//
#include <hip/hip_runtime.h>
#include <hip/hip_bf16.h>

typedef __attribute__((ext_vector_type(2))) float v2f;
typedef __attribute__((ext_vector_type(8))) float v8f;

constexpr int C_   = 64;
constexpr int O_   = 20;
constexpr int Dd   = 128;
constexpr int Hh   = 128;
constexpr int Ww   = 32;
constexpr int Nvox = Dd * Hh * Ww;   // 524288
constexpr float EPSF  = 1e-5f;
constexpr float THRES = 0.2f;

static __device__ __forceinline__ v8f wmma4(v2f a, v2f b, v8f c) {
  // V_WMMA_F32_16X16X4_F32: D = A(16x4) * B(4x16) + C(16x16)
  return __builtin_amdgcn_wmma_f32_16x16x4_f32(false, a, false, b, (short)0, c, false, false);
}

// order-preserving float<->uint for global atomic min/max
static __device__ __forceinline__ unsigned encf(float f) {
  unsigned u = __float_as_uint(f);
  return (u & 0x80000000u) ? ~u : (u | 0x80000000u);
}
static __device__ __forceinline__ float decf(unsigned u) {
  return (u & 0x80000000u) ? __uint_as_float(u ^ 0x80000000u) : __uint_as_float(~u);
}

// ---------------- init scratch (S1,S2 zero; minmax identity) ----------------
__global__ void k_init(float* S1, float* S2, unsigned* mnmx) {
  int t = blockIdx.x * 256 + threadIdx.x;
  if (t < C_ * O_) { S1[t] = 0.f; S2[t] = 0.f; }
  if (t < 2 * O_) mnmx[t] = (t & 1) ? 0u : 0xFFFFFFFFu;  // even=min slot, odd=max slot
}

// ---------------- per-voxel channel mean / max ----------------
__global__ void k_meanmax(const float* __restrict__ x, float* __restrict__ xmean,
                          float* __restrict__ xmax) {
  int v = blockIdx.x * 256 + threadIdx.x;
  float s = 0.f, m = -3.4e38f;
  #pragma unroll 8
  for (int c = 0; c < C_; ++c) {
    float val = x[c * Nvox + v];
    s += val;
    m = fmaxf(m, val);
  }
  xmean[v] = s * (1.f / C_);
  xmax[v]  = m;
}

// ---------------- logit GEMM (WMMA f32 16x16x4) + bias + softmax ----------------
// M = classes (20 -> two 16-tiles, zero-padded in LDS), N = voxels (16/wave,
// 128/block), K = channels 64.
__global__ void k_logit_softmax(const float* __restrict__ x, const float* __restrict__ cls_w,
                                const float* __restrict__ cls_b, float* __restrict__ logit,
                                float* __restrict__ masks) {
  __shared__ float xs[64 * 128];   // B operand: x tile [channel][voxel]
  __shared__ float aw[32 * 64];    // A operand: cls_w zero-padded to 32 rows
  const int tid = threadIdx.x;
  const int vb  = blockIdx.x * 128;

  // cooperative coalesced load of x-tile [64 channels x 128 voxels]
  #pragma unroll
  for (int r = 0; r < 8; ++r) {
    int f    = tid + 256 * r;          // float4 index (2048 total)
    int row  = f >> 5;                 // 32 float4 per row
    int col4 = (f & 31) << 2;
    *reinterpret_cast<float4*>(&xs[row * 128 + col4]) =
        *reinterpret_cast<const float4*>(&x[row * Nvox + vb + col4]);
  }
  // cls_w -> LDS, rows 20..31 zeroed (unconditional clamped load + select)
  #pragma unroll
  for (int r = 0; r < 8; ++r) {
    int idx = tid + 256 * r;           // 2048 total
    float w = cls_w[min(idx, O_ * C_ - 1)];
    aw[idx] = (idx < O_ * C_) ? w : 0.f;
  }
  __syncthreads();

  const int wave = tid >> 5, lane = tid & 31;
  const int hi = lane >> 4, n = lane & 15;
  const int wv0 = wave * 16;

  v8f c0 = {}; v8f c1 = {};
  for (int kb = 0; kb < 64; kb += 4) {
    int k0 = kb + 2 * hi;
    v2f a0, a1, b;
    // A: lane(0-15)=M row, regs+lane.hi = K   (16x4 f32 layout)
    a0.x = aw[n * C_ + k0];
    a0.y = aw[n * C_ + k0 + 1];
    a1.x = aw[(16 + n) * C_ + k0];
    a1.y = aw[(16 + n) * C_ + k0 + 1];
    // B: lane(0-15)=N col, regs+lane.hi = K   (4x16 f32 layout)
    b.x = xs[k0 * 128 + wv0 + n];
    b.y = xs[(k0 + 1) * 128 + wv0 + n];
    c0 = wmma4(a0, b, c0);
    c1 = wmma4(a1, b, c1);
  }
  __syncthreads();

  // stash logits: lg[class][local voxel]; C/D layout: M = j + 8*hi, N = n
  float* lg = xs;  // reuse LDS
  #pragma unroll
  for (int j = 0; j < 8; ++j) {
    int cls = j + 8 * hi;
    lg[cls * 128 + wv0 + n] = c0[j];
    int cls1 = 16 + cls;
    if (cls1 < O_) lg[cls1 * 128 + wv0 + n] = c1[j];
  }
  __syncthreads();

  if (tid < 128) {
    int v = vb + tid;
    float l[O_];
    float mx = -3.4e38f;
    #pragma unroll
    for (int o = 0; o < O_; ++o) {
      float raw = lg[o * 128 + tid] + cls_b[o];
      l[o] = raw;
      logit[o * Nvox + v] = raw;
      mx = fmaxf(mx, raw);
    }
    float sum = 0.f;
    #pragma unroll
    for (int o = 0; o < O_; ++o) { l[o] = __expf(l[o] - mx); sum += l[o]; }
    float inv = 1.f / sum;
    #pragma unroll
    for (int o = 0; o < O_; ++o) masks[o * Nvox + v] = l[o] * inv;
  }
}

// ---------------- per-class 7x7x7 conv on derived channels + sigmoid + heat minmax ----
// channels: [mask*xmean, mask*xmax, mask]; output s = sigmoid(conv)*mask -> tbuf
__global__ void k_conv(const float* __restrict__ masks, const float* __restrict__ xmean,
                       const float* __restrict__ xmax, const float* __restrict__ cfr_w,
                       float* __restrict__ tbuf, unsigned* __restrict__ mnmx) {
  constexpr int IZ = 10, IY = 10, IX = 38;      // 4+6, 4+6, 32+6
  __shared__ float sm[IZ * IY * IX];
  __shared__ float sa[IZ * IY * IX];
  __shared__ float sb[IZ * IY * IX];
  __shared__ float wt[3 * 343];
  __shared__ float redmn[256], redmx[256];

  const int cls = blockIdx.z;
  const int d0 = blockIdx.x * 4, h0 = blockIdx.y * 4;
  const int tid = threadIdx.x;
  const float* mptr = masks + cls * Nvox;

  for (int idx = tid; idx < IZ * IY * IX; idx += 256) {
    int iz = idx / (IY * IX);
    int rem = idx % (IY * IX);
    int iy = rem / IX, ix = rem % IX;
    int d = d0 + iz - 3, h = h0 + iy - 3, w = ix - 3;
    float m = 0.f, a = 0.f, b = 0.f;
    if (d >= 0 && d < Dd && h >= 0 && h < Hh && w >= 0 && w < Ww) {
      int v = (d * Hh + h) * Ww + w;
      m = mptr[v]; a = xmean[v]; b = xmax[v];
    }
    sm[idx] = m; sa[idx] = a; sb[idx] = b;
  }
  for (int idx = tid; idx < 3 * 343; idx += 256) wt[idx] = cfr_w[cls * 3 * 343 + idx];
  __syncthreads();

  float hmn = 3.4e38f, hmx = -3.4e38f;
  #pragma unroll
  for (int rep = 0; rep < 2; ++rep) {
    int o = tid + rep * 256;                    // 4x4x32 = 512 outputs
    int oz = o >> 7, orem = o & 127, oy = orem >> 5, ox = orem & 31;
    float acc = 0.f;
    for (int kz = 0; kz < 7; ++kz)
      for (int ky = 0; ky < 7; ++ky) {
        int base  = ((oz + kz) * IY + (oy + ky)) * IX + ox;
        int wbase = (kz * 7 + ky) * 7;
        #pragma unroll
        for (int kx = 0; kx < 7; ++kx) {
          float m = sm[base + kx];
          float w0 = wt[wbase + kx], w1 = wt[343 + wbase + kx], w2 = wt[686 + wbase + kx];
          acc += m * (w0 * sa[base + kx] + w1 * sb[base + kx] + w2);
        }
      }
    float att = 1.f / (1.f + __expf(-acc));
    int cb = ((oz + 3) * IY + (oy + 3)) * IX + ox + 3;
    float s = att * sm[cb];
    int vout = ((d0 + oz) * Hh + (h0 + oy)) * Ww + ox;
    tbuf[cls * Nvox + vout] = s;
    float heat = s * sa[cb];
    hmn = fminf(hmn, heat); hmx = fmaxf(hmx, heat);
  }

  redmn[tid] = hmn; redmx[tid] = hmx;
  __syncthreads();
  for (int off = 128; off > 0; off >>= 1) {
    if (tid < off) {
      redmn[tid] = fminf(redmn[tid], redmn[tid + off]);
      redmx[tid] = fmaxf(redmx[tid], redmx[tid + off]);
    }
    __syncthreads();
  }
  if (tid == 0) {
    atomicMin(&mnmx[2 * cls],     encf(redmn[0]));
    atomicMax(&mnmx[2 * cls + 1], encf(redmx[0]));
  }
}

// ---------------- region threshold: t = s * (norm(heat) > THRES) ----------------
__global__ void k_thresh(float* __restrict__ tbuf, const float* __restrict__ xmean,
                         const unsigned* __restrict__ mnmx) {
  int i = blockIdx.y;
  int v = blockIdx.x * 256 + threadIdx.x;
  float s = tbuf[i * Nvox + v];
  float heat = s * xmean[v];
  float mn = decf(mnmx[2 * i]), mx = decf(mnmx[2 * i + 1]);
  float norm = (heat - mn) / (mx - mn);
  tbuf[i * Nvox + v] = (norm > THRES) ? s : 0.f;
}

// ---------------- instance-norm stats: S1 = x*t^T, S2 = x^2*(t^2)^T (WMMA) ----------
// M = channel (4 tiles), N = class (2 tiles), K = voxels (chunked by 64 via LDS)
__global__ void k_stats(const float* __restrict__ x, const float* __restrict__ tbuf,
                        float* __restrict__ S1, float* __restrict__ S2) {
  __shared__ float xs[64 * 64];
  __shared__ float ts[32 * 64];
  const int tid = threadIdx.x, wave = tid >> 5, lane = tid & 31;
  const int hi = lane >> 4, n = lane & 15;
  const int mtile = wave & 3, sgrp = wave >> 2;   // sgrp 0 -> S1, 1 -> S2

  v8f acc0 = {}; v8f acc1 = {};
  for (int chunk = blockIdx.x; chunk < Nvox / 64; chunk += gridDim.x) {
    int vb = chunk * 64;
    __syncthreads();
    #pragma unroll
    for (int r = 0; r < 4; ++r) {
      int f = tid + 256 * r;                     // float4 idx (1024 total)
      int row = f >> 4, col4 = (f & 15) << 2;
      *reinterpret_cast<float4*>(&xs[row * 64 + col4]) =
          *reinterpret_cast<const float4*>(&x[row * Nvox + vb + col4]);
    }
    #pragma unroll
    for (int r = 0; r < 8; ++r) {
      int idx = tid + 256 * r;                   // 2048 total (32 rows x 64)
      int row = idx >> 6, col = idx & 63;
      float tv = tbuf[min(row, O_ - 1) * Nvox + vb + col];  // unconditional load
      ts[idx] = (row < O_) ? tv : 0.f;
    }
    __syncthreads();
    for (int kb = 0; kb < 64; kb += 4) {
      int k0 = kb + 2 * hi;
      int c = mtile * 16 + n;
      v2f a, b0, b1;
      a.x  = xs[c * 64 + k0];        a.y  = xs[c * 64 + k0 + 1];
      b0.x = ts[n * 64 + k0];        b0.y = ts[n * 64 + k0 + 1];
      b1.x = ts[(16 + n) * 64 + k0]; b1.y = ts[(16 + n) * 64 + k0 + 1];
      if (sgrp) {  // wave-uniform: square operands for E[y^2]
        a.x *= a.x; a.y *= a.y;
        b0.x *= b0.x; b0.y *= b0.y;
        b1.x *= b1.x; b1.y *= b1.y;
      }
      acc0 = wmma4(a, b0, acc0);
      acc1 = wmma4(a, b1, acc1);
    }
  }
  float* S = sgrp ? S2 : S1;
  #pragma unroll
  for (int j = 0; j < 8; ++j) {
    int c = mtile * 16 + j + 8 * hi;
    atomicAdd(&S[c * O_ + n], acc0[j]);
    int cl1 = 16 + n;
    if (cl1 < O_) atomicAdd(&S[c * O_ + cl1], acc1[j]);
  }
}

// ---------------- fold stats into R[c,i] = gamma*rsqrt(var+eps), Kc[c] ----------------
__global__ void k_finstats(const float* __restrict__ S1, const float* __restrict__ S2,
                           const float* __restrict__ gamma, const float* __restrict__ beta,
                           float* __restrict__ Rm, float* __restrict__ Kc) {
  int c = threadIdx.x;
  if (c < C_) {
    float g = gamma[c], be = beta[c];
    float kc = 0.f;
    for (int i = 0; i < O_; ++i) {
      float mean = S1[c * O_ + i] * (1.f / Nvox);
      float ex2  = S2[c * O_ + i] * (1.f / Nvox);
      float var  = ex2 - mean * mean;
      float rs   = rsqrtf(var + EPSF);
      Rm[c * O_ + i] = g * rs;
      kc += be - g * mean * rs;
    }
    Kc[c] = kc;
  }
}

// ---------------- final: out = relu(x * (R . t) + Kc) ----------------
__global__ void k_final(const float* __restrict__ x, const float* __restrict__ tbuf,
                        const float* __restrict__ Rm, const float* __restrict__ Kc,
                        float* __restrict__ outp) {
  __shared__ float sR[C_ * O_];
  __shared__ float sK[C_];
  int tid = threadIdx.x;
  for (int idx = tid; idx < C_ * O_; idx += 256) sR[idx] = Rm[idx];
  if (tid < C_) sK[tid] = Kc[tid];
  __syncthreads();
  int v = blockIdx.x * 256 + tid;
  float tv[O_];
  #pragma unroll
  for (int i = 0; i < O_; ++i) tv[i] = tbuf[i * Nvox + v];
  for (int c = 0; c < C_; ++c) {
    float u = 0.f;
    #pragma unroll
    for (int i = 0; i < O_; ++i) u += sR[c * O_ + i] * tv[i];
    float y = x[c * Nvox + v] * u + sK[c];
    outp[c * Nvox + v] = fmaxf(y, 0.f);
  }
}

extern "C" void kernel_launch(void* const* d_in, const int* in_sizes, int n_in,
                              void* d_out, int out_size, void* d_ws, size_t ws_size,
                              hipStream_t stream) {
  const float* x     = (const float*)d_in[0];
  const float* cls_w = (const float*)d_in[1];
  const float* cls_b = (const float*)d_in[2];
  const float* cfr_w = (const float*)d_in[3];
  const float* gamma = (const float*)d_in[4];
  const float* beta  = (const float*)d_in[5];

  float* out   = (float*)d_out;            // (1,64,D,H,W)
  float* logit = out + (size_t)C_ * Nvox;  // (1,20,D,H,W)

  float* ws    = (float*)d_ws;
  float* xmean = ws;
  float* xmax  = xmean + Nvox;
  float* masks = xmax + Nvox;
  float* tbuf  = masks + (size_t)O_ * Nvox;
  float* S1    = tbuf + (size_t)O_ * Nvox;
  float* S2    = S1 + C_ * O_;
  float* Rm    = S2 + C_ * O_;
  float* Kc    = Rm + C_ * O_;
  unsigned* mnmx = (unsigned*)(Kc + C_);

  k_init<<<5, 256, 0, stream>>>(S1, S2, mnmx);
  k_meanmax<<<Nvox / 256, 256, 0, stream>>>(x, xmean, xmax);
  k_logit_softmax<<<Nvox / 128, 256, 0, stream>>>(x, cls_w, cls_b, logit, masks);
  k_conv<<<dim3(Dd / 4, Hh / 4, O_), 256, 0, stream>>>(masks, xmean, xmax, cfr_w, tbuf, mnmx);
  k_thresh<<<dim3(Nvox / 256, O_), 256, 0, stream>>>(tbuf, xmean, mnmx);
  k_stats<<<512, 256, 0, stream>>>(x, tbuf, S1, S2);
  k_finstats<<<1, 64, 0, stream>>>(S1, S2, gamma, beta, Rm, Kc);
  k_final<<<Nvox / 256, 256, 0, stream>>>(x, tbuf, Rm, Kc, out);
}